// captioner_4793183502896
// MI455X (gfx1250) — compile-verified
//
#include <hip/hip_runtime.h>
#include <hip/hip_bf16.h>

// Problem constants
#define H      500
#define D_IN   512
#define D_OUT  100
#define STEPS  49

// Padded sizes (multiples of 16 for WMMA tiling)
#define MP   64    // 49 -> 64 rows (time steps)
#define HP   512   // 500 -> 512 hidden
#define NP   112   // 100 -> 112 outputs

// Rows of W_hh^T resident in LDS for the recurrence (144*512*4B = 288KB;
// + 2KB h vector = 290KB of the WGP's 320KB LDS). Must be multiple of 4.
#define ROWS_LDS 144

// Workspace layout (float offsets)
#define OFF_SEQ    0                         // seq_pad  [64 x 512]
#define OFF_WIHT   (OFF_SEQ   + MP*D_IN)     // W_ih^T   [512 x 512]
#define OFF_WHHT   (OFF_WIHT  + D_IN*HP)     // W_hh^T   [512 x 512]
#define OFF_WOUTT  (OFF_WHHT  + HP*HP)       // W_out^T  [512 x 112]
#define OFF_XPROJ  (OFF_WOUTT + HP*NP)       // xproj    [64 x 512]
#define OFF_HBUF   (OFF_XPROJ + MP*HP)       // h_t      [64 x 512]

typedef float v2f __attribute__((ext_vector_type(2)));
typedef float v8f __attribute__((ext_vector_type(8)));

// ---------------------------------------------------------------------------
// Prep: materialize seq (transpose of NCHW feature map), pre-transpose all
// weight matrices into zero-padded, coalescing-friendly layouts, zero Hbuf.
// ---------------------------------------------------------------------------
__global__ void prep_kernel(const float* __restrict__ x,
                            const float* __restrict__ W_ih,
                            const float* __restrict__ W_hh,
                            const float* __restrict__ W_out,
                            float* __restrict__ ws) {
    const int tid = blockIdx.x * blockDim.x + threadIdx.x;
    const int nth = gridDim.x * blockDim.x;

    // seq_pad[t][d] = x[0][d][t/7][t%7] = x[d*49 + t]   (t < 49), else 0
    for (int i = tid; i < MP * D_IN; i += nth) {
        int t = i / D_IN, d = i % D_IN;
        ws[OFF_SEQ + i] = (t < STEPS) ? x[d * STEPS + t] : 0.0f;
    }
    // WihT[d][h] = W_ih[h][d]  (h < 500), else 0
    for (int i = tid; i < D_IN * HP; i += nth) {
        int d = i / HP, h = i % HP;
        ws[OFF_WIHT + i] = (h < H) ? W_ih[h * D_IN + d] : 0.0f;
    }
    // WhhT[j][i2] = W_hh[i2][j]  (i2,j < 500), else 0
    for (int i = tid; i < HP * HP; i += nth) {
        int j = i / HP, i2 = i % HP;
        ws[OFF_WHHT + i] = (j < H && i2 < H) ? W_hh[i2 * H + j] : 0.0f;
    }
    // WoutT[j][n] = W_out[n][j]  (j < 500, n < 100), else 0
    for (int i = tid; i < HP * NP; i += nth) {
        int j = i / NP, n = i % NP;
        ws[OFF_WOUTT + i] = (j < H && n < D_OUT) ? W_out[n * H + j] : 0.0f;
    }
    // Zero Hbuf (rows >= 49 are read as padded A rows by the output GEMM)
    for (int i = tid; i < MP * HP; i += nth) {
        ws[OFF_HBUF + i] = 0.0f;
    }
}

// ---------------------------------------------------------------------------
// One 16x16 f32 WMMA tile: C = A[m0:m0+16, 0:K] * B[0:K, n0:n0+16]
// A row-major (lda), B row-major (ldb). K multiple of 4. EXEC all-ones
// (full wave, no divergence before/inside the loop).
// ISA 7.12.2 layouts:
//   A 16x4 : lane -> M = lane%16 ; VGPR v -> K = v + 2*(lane>=16)
//   B 4x16 : lane -> N = lane%16 ; VGPR v -> K = v + 2*(lane>=16)
//   C 16x16: lane -> N = lane%16 ; VGPR v -> M = v + 8*(lane>=16)
// ---------------------------------------------------------------------------
__device__ __forceinline__ v8f wmma_tile_f32(const float* __restrict__ A, int lda,
                                             const float* __restrict__ B, int ldb,
                                             int m0, int n0, int K) {
    const int lane = threadIdx.x & 31;
    const int r    = lane & 15;
    const int kh   = (lane >> 4) * 2;   // 0 or 2
    v8f c = {};
#pragma unroll 4
    for (int k0 = 0; k0 < K; k0 += 4) {
        const float* ap = A + (size_t)(m0 + r) * lda + (k0 + kh);
        v2f a; a.x = ap[0];   a.y = ap[1];          // contiguous -> b64 load
        const float* bp = B + (size_t)(k0 + kh) * ldb + (n0 + r);
        v2f b; b.x = bp[0];   b.y = bp[ldb];
        c = __builtin_amdgcn_wmma_f32_16x16x4_f32(false, a, false, b,
                                                  (short)0, c, false, false);
    }
    return c;
}

// ---------------------------------------------------------------------------
// xproj = seq_pad(64x512) @ WihT(512x512) + (b_ih + b_hh)
// 4 x 32 tiles, one wave per tile. 32 blocks x 128 threads (4 waves).
// ---------------------------------------------------------------------------
__global__ void xproj_gemm_kernel(const float* __restrict__ ws_seq,
                                  const float* __restrict__ ws_wiht,
                                  const float* __restrict__ b_ih,
                                  const float* __restrict__ b_hh,
                                  float* __restrict__ xproj) {
    const int wave = (blockIdx.x * blockDim.x + threadIdx.x) >> 5;  // 0..127
    const int mt = wave >> 5;        // 0..3
    const int nt = wave & 31;        // 0..31

    v8f c = wmma_tile_f32(ws_seq, D_IN, ws_wiht, HP, mt * 16, nt * 16, D_IN);

    const int lane = threadIdx.x & 31;
    const int r  = lane & 15;
    const int hi = lane >> 4;
    const int n  = nt * 16 + r;
    const float bias = (n < H) ? (b_ih[n] + b_hh[n]) : 0.0f;  // pad cols stay 0
#pragma unroll
    for (int v = 0; v < 8; ++v) {
        const int m = mt * 16 + hi * 8 + v;
        xproj[m * HP + n] = c[v] + bias;
    }
}

// ---------------------------------------------------------------------------
// Serial recurrence: h_{t+1} = tanh(xproj[t] + WhhT^T h_t), 49 steps.
// Single 512-thread workgroup (16 waves, wave32). CDNA5's 320KB/WGP LDS lets
// us pin 288KB (144 rows) of W_hh^T on-chip once, cutting per-step L2 traffic
// from 1MB to 736KB; LDS (256B/clk, conflict-free consecutive-lane reads)
// streams in parallel with L2. Four independent accumulators break the
// dependent-FMA chain so 4 waves/SIMD cover VALU latency.
// ---------------------------------------------------------------------------
__global__ void __launch_bounds__(HP)
recurrence_kernel(const float* __restrict__ xproj,
                  const float* __restrict__ whht,
                  const float* __restrict__ hidden0,
                  float* __restrict__ hbuf_out) {
    __shared__ float h_lds[HP];
    __shared__ float w_lds[ROWS_LDS * HP];      // 288KB of W_hh^T (rows 0..143)

    const int i = threadIdx.x;                  // 0..511: output column
    h_lds[i] = (i < H) ? hidden0[i] : 0.0f;

    // Cooperative vectorized stage-in: global_load_b128 + ds_store_b128
    {
        const float4* __restrict__ src = (const float4*)whht;
        float4* dst = (float4*)w_lds;
#pragma unroll
        for (int v = i; v < (ROWS_LDS * HP) / 4; v += HP) {
            dst[v] = src[v];
        }
    }
    __syncthreads();

    for (int t = 0; t < STEPS; ++t) {
        float acc0 = xproj[t * HP + i];
        float acc1 = 0.0f, acc2 = 0.0f, acc3 = 0.0f;

        // LDS-resident rows
#pragma unroll 4
        for (int j = 0; j < ROWS_LDS; j += 4) {
            acc0 = fmaf(w_lds[(j + 0) * HP + i], h_lds[j + 0], acc0);
            acc1 = fmaf(w_lds[(j + 1) * HP + i], h_lds[j + 1], acc1);
            acc2 = fmaf(w_lds[(j + 2) * HP + i], h_lds[j + 2], acc2);
            acc3 = fmaf(w_lds[(j + 3) * HP + i], h_lds[j + 3], acc3);
        }
        // L2-resident rows (coalesced 2KB rows; rows 500..511 are zero pad)
#pragma unroll 4
        for (int j = ROWS_LDS; j < HP; j += 4) {
            acc0 = fmaf(whht[(j + 0) * HP + i], h_lds[j + 0], acc0);
            acc1 = fmaf(whht[(j + 1) * HP + i], h_lds[j + 1], acc1);
            acc2 = fmaf(whht[(j + 2) * HP + i], h_lds[j + 2], acc2);
            acc3 = fmaf(whht[(j + 3) * HP + i], h_lds[j + 3], acc3);
        }

        const float hn = tanhf((acc0 + acc1) + (acc2 + acc3)); // pad: tanh(0)=0
        __syncthreads();                 // all reads of h_lds done
        h_lds[i] = hn;
        hbuf_out[t * HP + i] = hn;
        __syncthreads();
    }
}

// ---------------------------------------------------------------------------
// out[t][n] = tanh(Hbuf(64x512) @ WoutT(512x112) + b_out), stored [49 x 100].
// 4 x 7 tiles = 28 waves. 7 blocks x 128 threads.
// ---------------------------------------------------------------------------
__global__ void out_gemm_kernel(const float* __restrict__ hbuf,
                                const float* __restrict__ woutt,
                                const float* __restrict__ b_out,
                                float* __restrict__ out) {
    const int wave = (blockIdx.x * blockDim.x + threadIdx.x) >> 5;  // 0..27
    const int mt = wave / 7;
    const int nt = wave % 7;

    v8f c = wmma_tile_f32(hbuf, HP, woutt, NP, mt * 16, nt * 16, HP);

    const int lane = threadIdx.x & 31;
    const int r  = lane & 15;
    const int hi = lane >> 4;
    const int n  = nt * 16 + r;
#pragma unroll
    for (int v = 0; v < 8; ++v) {
        const int m = mt * 16 + hi * 8 + v;
        if (m < STEPS && n < D_OUT) {
            out[m * D_OUT + n] = tanhf(c[v] + b_out[n]);
        }
    }
}

// ---------------------------------------------------------------------------
// Host launcher. Inputs (setup_inputs order, all f32):
//   0: x [1,512,7,7]  1: hidden0 [1,500]  2: W_ih [500,512]  3: W_hh [500,500]
//   4: b_ih [500]     5: b_hh [500]       6: W_out [100,500] 7: b_out [100]
// Output: [1,49,100] f32 (4900 floats).
// ---------------------------------------------------------------------------
extern "C" void kernel_launch(void* const* d_in, const int* in_sizes, int n_in,
                              void* d_out, int out_size, void* d_ws, size_t ws_size,
                              hipStream_t stream) {
    const float* x       = (const float*)d_in[0];
    const float* hidden0 = (const float*)d_in[1];
    const float* W_ih    = (const float*)d_in[2];
    const float* W_hh    = (const float*)d_in[3];
    const float* b_ih    = (const float*)d_in[4];
    const float* b_hh    = (const float*)d_in[5];
    const float* W_out   = (const float*)d_in[6];
    const float* b_out   = (const float*)d_in[7];
    float* out = (float*)d_out;
    float* ws  = (float*)d_ws;

    prep_kernel<<<1024, 256, 0, stream>>>(x, W_ih, W_hh, W_out, ws);

    xproj_gemm_kernel<<<32, 128, 0, stream>>>(ws + OFF_SEQ, ws + OFF_WIHT,
                                              b_ih, b_hh, ws + OFF_XPROJ);

    recurrence_kernel<<<1, HP, 0, stream>>>(ws + OFF_XPROJ, ws + OFF_WHHT,
                                            hidden0, ws + OFF_HBUF);

    out_gemm_kernel<<<7, 128, 0, stream>>>(ws + OFF_HBUF, ws + OFF_WOUTT,
                                           b_out, out);
}